// Pointnet2Backbone_p_70781061038979
// MI455X (gfx1250) — compile-verified
//
#include <hip/hip_runtime.h>
#include <cstdint>

// ---------------------------------------------------------------------------
// Types for CDNA5 WMMA f32 16x16x4:
//   A (16x4 f32): 2 floats/lane. lanes 0-15: M=lane, K={0,1}; lanes 16-31: M=lane-16, K={2,3}
//   B (4x16 f32): 2 floats/lane. lanes 0-15: N=lane, K={0,1}; lanes 16-31: N=lane-16, K={2,3}
//   C/D (16x16):  8 floats/lane. vgpr r: lanes 0-15 -> M=r, N=lane; lanes 16-31 -> M=r+8
// ---------------------------------------------------------------------------
typedef float v2f __attribute__((ext_vector_type(2)));
typedef float v8f __attribute__((ext_vector_type(8)));

__device__ __forceinline__ v8f wmma4(v2f a, v2f b, v8f c) {
  return __builtin_amdgcn_wmma_f32_16x16x4_f32(false, a, false, b, (short)0, c,
                                               false, false);
}

// ---------------------------------------------------------------------------
// CDNA5 async global->LDS staging (GLOBAL_LOAD_ASYNC_TO_LDS_B128, ASYNCcnt).
// Probe-derived signature: (int4 __device__*, int4 __shared__*, imm offset,
// imm cpol). Guarded so the file still compiles if builtins are absent.
// ---------------------------------------------------------------------------
#if defined(__has_builtin)
#if __has_builtin(__builtin_amdgcn_global_load_async_to_lds_b128) && \
    __has_builtin(__builtin_amdgcn_s_wait_asynccnt)
#define USE_ASYNC_LDS 1
#endif
#endif
#ifndef USE_ASYNC_LDS
#define USE_ASYNC_LDS 0
#endif

#if USE_ASYNC_LDS
typedef int v4i_vs __attribute__((vector_size(4 * sizeof(int))));
typedef __attribute__((address_space(1))) v4i_vs* gv4_t;
typedef __attribute__((address_space(3))) v4i_vs* lv4_t;
#endif

// Copy nfloats (multiple of 4, both pointers 16B-aligned) global -> LDS.
// All 256 threads participate; follow with async_wait_lds() + __syncthreads().
__device__ __forceinline__ void stage_to_lds(const float* __restrict__ g,
                                             float* l, int nfloats, int tid) {
#if USE_ASYNC_LDS
  for (int t = tid * 4; t < nfloats; t += 256 * 4)
    __builtin_amdgcn_global_load_async_to_lds_b128(
        (gv4_t)(unsigned long long)(g + t), (lv4_t)(void*)(l + t), 0, 0);
#else
  for (int t = tid * 4; t < nfloats; t += 256 * 4) {
    l[t + 0] = g[t + 0];
    l[t + 1] = g[t + 1];
    l[t + 2] = g[t + 2];
    l[t + 3] = g[t + 3];
  }
#endif
}

__device__ __forceinline__ void async_wait_lds() {
#if USE_ASYNC_LDS
  __builtin_amdgcn_s_wait_asynccnt(0);
#endif
}

#define NB 8
#define NHALF 10000

// ---------------------------------------------------------------------------
// Split (8,20000,4) into even/odd halves: xyz (B,10000,3) + feat (B,10000,1)
// ---------------------------------------------------------------------------
__global__ void split_kernel(const float* __restrict__ pc,
                             float* __restrict__ xa, float* __restrict__ fa,
                             float* __restrict__ xb, float* __restrict__ fb) {
  int t = blockIdx.x * blockDim.x + threadIdx.x;
  if (t >= NB * NHALF) return;
  int b = t / NHALF, i = t - b * NHALF;
  const float* pa = pc + ((size_t)b * 2 * NHALF + 2 * i) * 4;
  const float* pb = pa + 4;
  xa[(size_t)t * 3 + 0] = pa[0];
  xa[(size_t)t * 3 + 1] = pa[1];
  xa[(size_t)t * 3 + 2] = pa[2];
  fa[t] = pa[3];
  xb[(size_t)t * 3 + 0] = pb[0];
  xb[(size_t)t * 3 + 1] = pb[1];
  xb[(size_t)t * 3 + 2] = pb[2];
  fb[t] = pb[3];
}

// ---------------------------------------------------------------------------
// Farthest point sampling. One 512-thread workgroup per batch.
// Matches jax scan: emits current far (starts at 0), then updates dists &
// takes argmax (first-index tie break).
// ---------------------------------------------------------------------------
#define FPS_T 512
#define FPS_MAXP 20  // ceil(10000/512)

__global__ __launch_bounds__(FPS_T) void fps_kernel(const float* __restrict__ xyz,
                                                    int N, int npoint,
                                                    int* __restrict__ inds) {
  __shared__ float rd[16];
  __shared__ int ri[16];
  __shared__ int sfar;
  const int b = blockIdx.x;
  const float* X = xyz + (size_t)b * N * 3;
  float ld[FPS_MAXP];
#pragma unroll
  for (int p = 0; p < FPS_MAXP; ++p) ld[p] = 1e10f;
  const int lane = threadIdx.x & 31, wid = threadIdx.x >> 5;
  const int nw = blockDim.x >> 5;
  int far = 0;
  for (int it = 0; it < npoint; ++it) {
    if (threadIdx.x == 0) inds[(size_t)b * npoint + it] = far;
    if (it == npoint - 1) break;
    float fx = X[(size_t)far * 3], fy = X[(size_t)far * 3 + 1],
          fz = X[(size_t)far * 3 + 2];
    float bd = -1.f;
    int bi = 0x7fffffff;
    int p = 0;
    for (int j = threadIdx.x; j < N; j += FPS_T, ++p) {
      float dx = X[(size_t)j * 3] - fx;
      float dy = X[(size_t)j * 3 + 1] - fy;
      float dz = X[(size_t)j * 3 + 2] - fz;
      float d = dx * dx + dy * dy + dz * dz;
      float nd = ld[p] < d ? ld[p] : d;
      ld[p] = nd;
      if (nd > bd) { bd = nd; bi = j; }
    }
    for (int off = 16; off > 0; off >>= 1) {
      float od = __shfl_down(bd, off);
      int oi = __shfl_down(bi, off);
      if (od > bd || (od == bd && oi < bi)) { bd = od; bi = oi; }
    }
    if (lane == 0) { rd[wid] = bd; ri[wid] = bi; }
    __syncthreads();
    if (threadIdx.x == 0) {
      float gb = rd[0];
      int gi = ri[0];
      for (int w = 1; w < nw; ++w) {
        float od = rd[w];
        int oi = ri[w];
        if (od > gb || (od == gb && oi < gi)) { gb = od; gi = oi; }
      }
      sfar = gi;
    }
    __syncthreads();
    far = sfar;
  }
}

// ---------------------------------------------------------------------------
// Gather selected centers: out[b,m,:] = xyz[b, inds[b,m], :]
// ---------------------------------------------------------------------------
__global__ void gather_xyz_kernel(const float* __restrict__ xyz,
                                  const int* __restrict__ inds, int N, int M,
                                  float* __restrict__ out, int total) {
  int t = blockIdx.x * blockDim.x + threadIdx.x;
  if (t >= total) return;
  int b = t / M;
  int i = inds[t];
  const float* s = xyz + ((size_t)b * N + i) * 3;
  out[(size_t)t * 3 + 0] = s[0];
  out[(size_t)t * 3 + 1] = s[1];
  out[(size_t)t * 3 + 2] = s[2];
}

// ---------------------------------------------------------------------------
// Ball query: one wave32 per group; collects first `ns` in-radius indices in
// ascending order (== jnp.sort of keys), pads with first hit (or N-1 if none).
// ---------------------------------------------------------------------------
__global__ void ball_query_kernel(const float* __restrict__ xyz, int N,
                                  const float* __restrict__ nxyz, int M,
                                  float r2, int ns, int* __restrict__ outidx,
                                  int groups) {
  const int lane = threadIdx.x & 31;
  const int g = blockIdx.x * (blockDim.x >> 5) + (threadIdx.x >> 5);
  if (g >= groups) return;
  const int b = g / M;
  const float* X = xyz + (size_t)b * N * 3;
  const float cx = nxyz[(size_t)g * 3], cy = nxyz[(size_t)g * 3 + 1],
              cz = nxyz[(size_t)g * 3 + 2];
  int* out = outidx + (size_t)g * ns;
  int cnt = 0;
  int firstIdx = N - 1;
  bool havefirst = false;
  for (int base = 0; base < N && cnt < ns; base += 32) {
    int j = base + lane;
    bool in = false;
    if (j < N) {
      float dx = X[(size_t)j * 3] - cx;
      float dy = X[(size_t)j * 3 + 1] - cy;
      float dz = X[(size_t)j * 3 + 2] - cz;
      in = (dx * dx + dy * dy + dz * dz) < r2;
    }
    unsigned mask = (unsigned)__ballot(in);  // wave32: low 32 bits
    if (!havefirst && mask) {
      firstIdx = base + (__ffs(mask) - 1);
      havefirst = true;
    }
    unsigned lt = (lane == 0) ? 0u : (0xffffffffu >> (32 - lane));
    int pos = __popc(mask & lt);
    if (in && (cnt + pos) < ns) out[cnt + pos] = j;
    cnt += __popc(mask);
  }
  int c = cnt < ns ? cnt : ns;
  if (c == 0) {
    for (int t = lane; t < ns; t += 32) out[t] = N - 1;
  } else {
    for (int t = lane; t < ns; t += 32)
      if (t >= c) out[t] = firstIdx;
  }
}

// ---------------------------------------------------------------------------
// Fused SA group MLP: gather (nsample x C0) into LDS, run 3 fc+relu layers
// with V_WMMA_F32_16X16X4_F32, max-pool over samples.
// One workgroup (8 waves) per group. LDS pool: slotA (4224 f) + slotB (8192 f)
// + W K-slab (16x128, async-staged). ~58 KB static LDS.
// ---------------------------------------------------------------------------
__device__ void sa_layer(const float* Xin, int ldin, int cin, float* Xout,
                         int ldout, int S, const float* __restrict__ Wg,
                         const float* __restrict__ bias, int cout, int tid,
                         float* Wslab) {
  const int cinp = (cin + 3) & ~3;
  const int ntn = cout >> 4;
  const int ntiles = (S >> 4) * ntn;  // multiple of 8 for all our shapes
  const int nrep = ntiles >> 3;
  const int wid = tid >> 5, lane = tid & 31;
  const int lrow = lane & 15;
  const int ko = (lane < 16) ? 0 : 2;
  for (int rep = 0; rep < nrep; ++rep) {
    const int ti = (rep << 3) + wid;
    const int tm = ti / ntn, tn = ti - tm * ntn;
    v8f acc;
#pragma unroll
    for (int r = 0; r < 8; ++r) acc[r] = 0.f;
    const float* xr = Xin + (size_t)((tm << 4) + lrow) * ldin;
    for (int k0 = 0; k0 < cinp; k0 += 16) {
      int kspan = cinp - k0;
      if (kspan > 16) kspan = 16;
      int nvalid = cin - k0;
      if (nvalid > 16) nvalid = 16;
      // async stage of up-to-16 K-rows of W (contiguous in global)
      stage_to_lds(Wg + (size_t)k0 * cout, Wslab, nvalid * cout, tid);
      // zero-fill K-pad rows (disjoint LDS region, ordered by the barrier)
      for (int t = tid; t < (kspan - nvalid) * cout; t += 256)
        Wslab[nvalid * cout + t] = 0.f;
      async_wait_lds();
      __syncthreads();
      const float* wc = Wslab + (tn << 4) + lrow;
      if (kspan == 16) {
#pragma unroll
        for (int k = 0; k < 16; k += 4) {
          v2f a = {xr[k0 + k + ko], xr[k0 + k + ko + 1]};
          v2f bb = {wc[(k + ko) * cout], wc[(k + ko + 1) * cout]};
          acc = wmma4(a, bb, acc);
        }
      } else {
        for (int k = 0; k < kspan; k += 4) {
          v2f a = {xr[k0 + k + ko], xr[k0 + k + ko + 1]};
          v2f bb = {wc[(k + ko) * cout], wc[(k + ko + 1) * cout]};
          acc = wmma4(a, bb, acc);
        }
      }
      __syncthreads();
    }
    const int col = (tn << 4) + lrow;
    const float bv = bias[col];
    const int mbase = (tm << 4) + ((lane < 16) ? 0 : 8);
#pragma unroll
    for (int r = 0; r < 8; ++r) {
      float v = acc[r] + bv;
      Xout[(size_t)(mbase + r) * ldout + col] = v > 0.f ? v : 0.f;
    }
    __syncthreads();
  }
}

__global__ __launch_bounds__(256) void sa_group_mlp_kernel(
    const float* __restrict__ xyz, const float* __restrict__ feats, int Cf,
    int N, const float* __restrict__ nxyz, const int* __restrict__ gidx, int M,
    int S, float inv_r, const float* __restrict__ w1,
    const float* __restrict__ b1, int c1, const float* __restrict__ w2,
    const float* __restrict__ b2, int c2, const float* __restrict__ w3,
    const float* __restrict__ b3, int c3, float* __restrict__ outf) {
  __shared__ __align__(16) float pool[12416];  // slotA: 0..4223, slotB: 4224+
  __shared__ __align__(16) float Wslab[2048];  // 16 x up-to-128
  float* bufA = pool;
  float* bufB = pool + 4224;
  const int tid = threadIdx.x;
  const int g = blockIdx.x;
  const int b = g / M;
  const int C0 = 3 + Cf;
  const int C0p = (C0 + 3) & ~3;
  const float cx = nxyz[(size_t)g * 3 + 0];
  const float cy = nxyz[(size_t)g * 3 + 1];
  const float cz = nxyz[(size_t)g * 3 + 2];
  const int* gi = gidx + (size_t)g * S;
  const float* Xg = xyz + (size_t)b * N * 3;
  const float* Fg = feats + (size_t)b * N * Cf;
  // gather grouped input: cols 0..2 = (p - center)/radius, cols 3.. = feats
  for (int t = tid; t < S * C0p; t += 256) {
    int s = t / C0p, c = t - s * C0p;
    int p = gi[s];
    float v;
    if (c == 0) v = (Xg[(size_t)p * 3] - cx) * inv_r;
    else if (c == 1) v = (Xg[(size_t)p * 3 + 1] - cy) * inv_r;
    else if (c == 2) v = (Xg[(size_t)p * 3 + 2] - cz) * inv_r;
    else if (c < C0) v = Fg[(size_t)p * Cf + (c - 3)];
    else v = 0.f;  // K-pad column
    bufA[s * C0p + c] = v;
  }
  __syncthreads();
  sa_layer(bufA, C0p, C0, bufB, c1, S, w1, b1, c1, tid, Wslab);
  sa_layer(bufB, c1, c1, bufA, c2, S, w2, b2, c2, tid, Wslab);
  sa_layer(bufA, c2, c2, bufB, c3, S, w3, b3, c3, tid, Wslab);
  // max-pool over samples
  for (int n = tid; n < c3; n += 256) {
    float m = bufB[n];
    for (int s = 1; s < S; ++s) m = fmaxf(m, bufB[s * c3 + n]);
    outf[(size_t)g * c3 + n] = m;
  }
}

// ---------------------------------------------------------------------------
// Concat two (B,Mh,C) along point axis -> (B,2Mh,C)
// ---------------------------------------------------------------------------
__global__ void concat2_kernel(const float* __restrict__ A,
                               const float* __restrict__ Bp, int Mh, int C,
                               float* __restrict__ out, int total) {
  int t = blockIdx.x * blockDim.x + threadIdx.x;
  if (t >= total) return;
  int c = t % C;
  int pm = t / C;
  int m = pm % (2 * Mh);
  int b = pm / (2 * Mh);
  out[t] = (m < Mh) ? A[((size_t)b * Mh + m) * C + c]
                    : Bp[((size_t)b * Mh + (m - Mh)) * C + c];
}

// ---------------------------------------------------------------------------
// 3-NN (thread per query point) with first-index tie break (== top_k).
// ---------------------------------------------------------------------------
__global__ void knn3_kernel(const float* __restrict__ x1, int M1,
                            const float* __restrict__ x2, int M2,
                            int* __restrict__ kidx, float* __restrict__ kw,
                            int total) {
  int t = blockIdx.x * blockDim.x + threadIdx.x;
  if (t >= total) return;
  int b = t / M1;
  float px = x1[(size_t)t * 3], py = x1[(size_t)t * 3 + 1],
        pz = x1[(size_t)t * 3 + 2];
  const float* X2 = x2 + (size_t)b * M2 * 3;
  float d0 = 3.4e38f, d1 = 3.4e38f, d2v = 3.4e38f;
  int i0 = 0, i1 = 0, i2 = 0;
  for (int j = 0; j < M2; ++j) {
    float dx = X2[(size_t)j * 3] - px;
    float dy = X2[(size_t)j * 3 + 1] - py;
    float dz = X2[(size_t)j * 3 + 2] - pz;
    float d = dx * dx + dy * dy + dz * dz;
    if (d < d0) { d2v = d1; i2 = i1; d1 = d0; i1 = i0; d0 = d; i0 = j; }
    else if (d < d1) { d2v = d1; i2 = i1; d1 = d; i1 = j; }
    else if (d < d2v) { d2v = d; i2 = j; }
  }
  float w0 = 1.f / (sqrtf(fmaxf(d0, 0.f)) + 1e-8f);
  float w1 = 1.f / (sqrtf(fmaxf(d1, 0.f)) + 1e-8f);
  float w2 = 1.f / (sqrtf(fmaxf(d2v, 0.f)) + 1e-8f);
  float s = w0 + w1 + w2;
  kidx[(size_t)t * 3 + 0] = i0;
  kidx[(size_t)t * 3 + 1] = i1;
  kidx[(size_t)t * 3 + 2] = i2;
  kw[(size_t)t * 3 + 0] = w0 / s;
  kw[(size_t)t * 3 + 1] = w1 / s;
  kw[(size_t)t * 3 + 2] = w2 / s;
}

// ---------------------------------------------------------------------------
// FP feature build: x = [interp3(f2), f1]   (block per point, coalesced cols)
// ---------------------------------------------------------------------------
__global__ void fp_concat_kernel(const float* __restrict__ f2, int C2, int M2,
                                 const float* __restrict__ f1, int C1, int M1,
                                 const int* __restrict__ kidx,
                                 const float* __restrict__ kw,
                                 float* __restrict__ xout) {
  int p = blockIdx.x;
  int b = p / M1;
  int i0 = kidx[(size_t)p * 3], i1 = kidx[(size_t)p * 3 + 1],
      i2 = kidx[(size_t)p * 3 + 2];
  float w0 = kw[(size_t)p * 3], w1 = kw[(size_t)p * 3 + 1],
        w2 = kw[(size_t)p * 3 + 2];
  const float* F2 = f2 + (size_t)b * M2 * C2;
  float* o = xout + (size_t)p * (C2 + C1);
  for (int c = threadIdx.x; c < C2; c += blockDim.x)
    o[c] = w0 * F2[(size_t)i0 * C2 + c] + w1 * F2[(size_t)i1 * C2 + c] +
           w2 * F2[(size_t)i2 * C2 + c];
  const float* F1 = f1 + (size_t)p * C1;
  for (int c = threadIdx.x; c < C1; c += blockDim.x) o[C2 + c] = F1[c];
}

// ---------------------------------------------------------------------------
// WMMA GEMM: Y[R x Nc] = relu(X[R x Cin] @ W[Cin x Nc] + bias).
// Block tile 32x64, 8 waves x one 16x16 tile, X async-staged to LDS, W from L2.
// R % 32 == 0, Nc % 64 == 0, Cin % 4 == 0, Cin <= 388.
// ---------------------------------------------------------------------------
#define GEMM_KMAX 388
__global__ __launch_bounds__(256) void gemm_bias_relu_kernel(
    const float* __restrict__ X, int Cin, const float* __restrict__ Wt,
    const float* __restrict__ bias, int Nc, float* __restrict__ Y) {
  __shared__ __align__(16) float Xs[32 * GEMM_KMAX];
  const int tid = threadIdx.x;
  const size_t rb = (size_t)blockIdx.x * 32;
  const int cb = blockIdx.y * 64;
  // 32 consecutive rows of X are one contiguous block: async copy to LDS.
  stage_to_lds(X + rb * Cin, Xs, 32 * Cin, tid);
  async_wait_lds();
  __syncthreads();
  const int wid = tid >> 5, lane = tid & 31;
  const int tm = wid >> 2, tn = wid & 3;
  const int lrow = lane & 15;
  const int ko = (lane < 16) ? 0 : 2;
  const float* xr = Xs + (size_t)((tm << 4) + lrow) * Cin;
  const int col = cb + (tn << 4) + lrow;
  v8f acc;
#pragma unroll
  for (int r = 0; r < 8; ++r) acc[r] = 0.f;
#pragma unroll 4
  for (int k = 0; k < Cin; k += 4) {
    v2f a = {xr[k + ko], xr[k + ko + 1]};
    v2f bb = {Wt[(size_t)(k + ko) * Nc + col],
              Wt[(size_t)(k + ko + 1) * Nc + col]};
    acc = wmma4(a, bb, acc);
  }
  const float bv = bias[col];
  const size_t mbase = rb + (tm << 4) + ((lane < 16) ? 0 : 8);
#pragma unroll
  for (int r = 0; r < 8; ++r) {
    float v = acc[r] + bv;
    Y[(mbase + r) * Nc + col] = v > 0.f ? v : 0.f;
  }
}

__global__ void i2f_kernel(const int* __restrict__ in, float* __restrict__ out,
                           int n) {
  int t = blockIdx.x * blockDim.x + threadIdx.x;
  if (t < n) out[t] = (float)in[t];
}

// ---------------------------------------------------------------------------
// Host orchestration
// ---------------------------------------------------------------------------
static void run_sa_stage(hipStream_t stream, const float* xyz,
                         const float* feats, int Cf, int N, int M, float radius,
                         int S, const float* w1, const float* b1, int c1,
                         const float* w2, const float* b2, int c2,
                         const float* w3, const float* b3, int c3, int* inds,
                         float* nxyz, int* bidx, float* outf) {
  fps_kernel<<<NB, FPS_T, 0, stream>>>(xyz, N, M, inds);
  int total = NB * M;
  gather_xyz_kernel<<<(total + 255) / 256, 256, 0, stream>>>(xyz, inds, N, M,
                                                             nxyz, total);
  ball_query_kernel<<<(total + 7) / 8, 256, 0, stream>>>(
      xyz, N, nxyz, M, radius * radius, S, bidx, total);
  sa_group_mlp_kernel<<<total, 256, 0, stream>>>(
      xyz, feats, Cf, N, nxyz, bidx, M, S, 1.0f / radius, w1, b1, c1, w2, b2,
      c2, w3, b3, c3, outf);
}

extern "C" void kernel_launch(void* const* d_in, const int* in_sizes, int n_in,
                              void* d_out, int out_size, void* d_ws,
                              size_t ws_size, hipStream_t stream) {
  (void)in_sizes; (void)n_in; (void)out_size; (void)ws_size;
  const float* pc = (const float*)d_in[0];
  const float* saw[4][3];
  const float* sab[4][3];
  int idx = 1;
  for (int s = 0; s < 4; ++s)
    for (int l = 0; l < 3; ++l) {
      saw[s][l] = (const float*)d_in[idx++];
      sab[s][l] = (const float*)d_in[idx++];
    }
  const float* fpw[2][2];
  const float* fpb[2][2];
  for (int s = 0; s < 2; ++s)
    for (int l = 0; l < 2; ++l) {
      fpw[s][l] = (const float*)d_in[idx++];
      fpb[s][l] = (const float*)d_in[idx++];
    }

  char* wsb = (char*)d_ws;
  size_t off = 0;
  auto allocf = [&](size_t n) -> float* {
    float* p = (float*)(wsb + off);
    off += ((n * sizeof(float) + 255) & ~(size_t)255);
    return p;
  };
  auto alloci = [&](size_t n) -> int* {
    int* p = (int*)(wsb + off);
    off += ((n * sizeof(int) + 255) & ~(size_t)255);
    return p;
  };

  float *xyz0[2], *f0[2], *xyz1[2], *f1[2], *xyz2[2], *f2[2], *xyz3[2], *f3[2],
      *xyz4[2], *f4[2];
  int *inds1[2], *indsT[2], *bidx1[2], *bidx2[2], *bidx3[2], *bidx4[2];
  for (int h = 0; h < 2; ++h) {
    xyz0[h] = allocf((size_t)NB * NHALF * 3);
    f0[h] = allocf((size_t)NB * NHALF);
    xyz1[h] = allocf((size_t)NB * 1024 * 3);
    f1[h] = allocf((size_t)NB * 1024 * 128);
    xyz2[h] = allocf((size_t)NB * 512 * 3);
    f2[h] = allocf((size_t)NB * 512 * 128);
    xyz3[h] = allocf((size_t)NB * 256 * 3);
    f3[h] = allocf((size_t)NB * 256 * 128);
    xyz4[h] = allocf((size_t)NB * 128 * 3);
    f4[h] = allocf((size_t)NB * 128 * 128);
    inds1[h] = alloci((size_t)NB * 1024);
    indsT[h] = alloci((size_t)NB * 1024);
    bidx1[h] = alloci((size_t)NB * 1024 * 64);
    bidx2[h] = alloci((size_t)NB * 512 * 32);
    bidx3[h] = alloci((size_t)NB * 256 * 16);
    bidx4[h] = alloci((size_t)NB * 128 * 16);
  }

  split_kernel<<<(NB * NHALF + 255) / 256, 256, 0, stream>>>(
      pc, xyz0[0], f0[0], xyz0[1], f0[1]);

  for (int h = 0; h < 2; ++h) {
    run_sa_stage(stream, xyz0[h], f0[h], 1, NHALF, 1024, 0.2f, 64, saw[0][0],
                 sab[0][0], 64, saw[0][1], sab[0][1], 64, saw[0][2], sab[0][2],
                 128, inds1[h], xyz1[h], bidx1[h], f1[h]);
    run_sa_stage(stream, xyz1[h], f1[h], 128, 1024, 512, 0.4f, 32, saw[1][0],
                 sab[1][0], 128, saw[1][1], sab[1][1], 128, saw[1][2],
                 sab[1][2], 128, indsT[h], xyz2[h], bidx2[h], f2[h]);
    run_sa_stage(stream, xyz2[h], f2[h], 128, 512, 256, 0.8f, 16, saw[2][0],
                 sab[2][0], 128, saw[2][1], sab[2][1], 128, saw[2][2],
                 sab[2][2], 128, indsT[h], xyz3[h], bidx3[h], f3[h]);
    run_sa_stage(stream, xyz3[h], f3[h], 128, 256, 128, 1.2f, 16, saw[3][0],
                 sab[3][0], 128, saw[3][1], sab[3][1], 128, saw[3][2],
                 sab[3][2], 128, indsT[h], xyz4[h], bidx4[h], f4[h]);
  }

  // concat halves along the point axis
  float* s4x = allocf((size_t)NB * 256 * 3);
  float* s4f = allocf((size_t)NB * 256 * 128);
  float* s3x = allocf((size_t)NB * 512 * 3);
  float* s3f = allocf((size_t)NB * 512 * 128);
  float* s2x = allocf((size_t)NB * 1024 * 3);
  float* s2f = allocf((size_t)NB * 1024 * 128);
  {
    int t;
    t = NB * 256 * 3;
    concat2_kernel<<<(t + 255) / 256, 256, 0, stream>>>(xyz4[0], xyz4[1], 128, 3, s4x, t);
    t = NB * 256 * 128;
    concat2_kernel<<<(t + 255) / 256, 256, 0, stream>>>(f4[0], f4[1], 128, 128, s4f, t);
    t = NB * 512 * 3;
    concat2_kernel<<<(t + 255) / 256, 256, 0, stream>>>(xyz3[0], xyz3[1], 256, 3, s3x, t);
    t = NB * 512 * 128;
    concat2_kernel<<<(t + 255) / 256, 256, 0, stream>>>(f3[0], f3[1], 256, 128, s3f, t);
    t = NB * 1024 * 3;
    concat2_kernel<<<(t + 255) / 256, 256, 0, stream>>>(xyz2[0], xyz2[1], 512, 3, s2x, t);
    t = NB * 1024 * 128;
    concat2_kernel<<<(t + 255) / 256, 256, 0, stream>>>(f2[0], f2[1], 512, 128, s2f, t);
  }

  // FP1: interp s4f onto s3x, concat s3f, 2x (fc 256 + relu)
  int* k3i = alloci((size_t)NB * 512 * 3);
  float* k3w = allocf((size_t)NB * 512 * 3);
  knn3_kernel<<<(NB * 512 + 255) / 256, 256, 0, stream>>>(s3x, 512, s4x, 256,
                                                          k3i, k3w, NB * 512);
  float* xb1 = allocf((size_t)NB * 512 * 256);
  fp_concat_kernel<<<NB * 512, 256, 0, stream>>>(s4f, 128, 256, s3f, 128, 512,
                                                 k3i, k3w, xb1);
  float* h1 = allocf((size_t)NB * 512 * 256);
  float* fp1f = allocf((size_t)NB * 512 * 256);
  dim3 g1(NB * 512 / 32, 256 / 64);
  gemm_bias_relu_kernel<<<g1, 256, 0, stream>>>(xb1, 256, fpw[0][0], fpb[0][0], 256, h1);
  gemm_bias_relu_kernel<<<g1, 256, 0, stream>>>(h1, 256, fpw[0][1], fpb[0][1], 256, fp1f);

  // FP2: interp fp1f onto s2x, concat s2f, 2x (fc 256 + relu) -> d_out
  int* k2i = alloci((size_t)NB * 1024 * 3);
  float* k2w = allocf((size_t)NB * 1024 * 3);
  knn3_kernel<<<(NB * 1024 + 255) / 256, 256, 0, stream>>>(s2x, 1024, s3x, 512,
                                                           k2i, k2w, NB * 1024);
  float* xb2 = allocf((size_t)NB * 1024 * 384);
  fp_concat_kernel<<<NB * 1024, 256, 0, stream>>>(fp1f, 256, 512, s2f, 128,
                                                  1024, k2i, k2w, xb2);
  float* h2 = allocf((size_t)NB * 1024 * 256);
  float* outf = (float*)d_out;
  dim3 g2(NB * 1024 / 32, 256 / 64);
  gemm_bias_relu_kernel<<<g2, 256, 0, stream>>>(xb2, 384, fpw[1][0], fpb[1][0], 256, h2);
  gemm_bias_relu_kernel<<<g2, 256, 0, stream>>>(h2, 256, fpw[1][1], fpb[1][1], 256, outf);

  // outputs 2 & 3: s2x coords and level-1 FPS indices of half a
  (void)hipMemcpyAsync(outf + (size_t)NB * 1024 * 256, s2x,
                       (size_t)NB * 1024 * 3 * sizeof(float),
                       hipMemcpyDeviceToDevice, stream);
  i2f_kernel<<<(NB * 1024 + 255) / 256, 256, 0, stream>>>(
      inds1[0], outf + (size_t)NB * 1024 * 256 + (size_t)NB * 1024 * 3,
      NB * 1024);
}